// DNN_24464133718540
// MI455X (gfx1250) — compile-verified
//
#include <hip/hip_runtime.h>
#include <hip/hip_bf16.h>

typedef __attribute__((ext_vector_type(2))) float v2f;
typedef __attribute__((ext_vector_type(8))) float v8f;

#define BB 16384
#define FF 45
#define VV 10000
#define DD 256
#define HH 32

// ---------------------------------------------------------------------------
// Kernel 1: fold the linear MLP.
//   w  = W1 (256x32) @ W2 (32x1)          -> wc[0..255]
//   c  = F * (b1 . W2 + b2)               -> wc[256]
// Done with V_WMMA_F32_16X16X4_F32: 16 waves, one 16-row M tile per wave,
// K accumulated 4 at a time. W2 is broadcast across all 16 N columns, so
// every column of D equals the w-tile; we extract column N=0 (lanes 0 / 16).
//
// f32 16x4 A layout (ISA 7.12.2): lane<16 holds M=lane, VGPR j -> K=j;
// lane>=16 holds M=lane-16, VGPR j -> K=j+2.  B (4x16) mirrors: VGPR j,
// lane-half hi -> row K = j + 2*hi, N = lane&15 (irrelevant under broadcast).
// C/D: VGPR r, lanes 0-15 -> M=r, lanes 16-31 -> M=r+8.
// ---------------------------------------------------------------------------
__global__ __launch_bounds__(512) void fold_mlp_kernel(
    const float* __restrict__ W1, const float* __restrict__ b1,
    const float* __restrict__ W2, const float* __restrict__ b2,
    float* __restrict__ wc)
{
    const int lane = threadIdx.x & 31;
    const int wave = threadIdx.x >> 5;      // M tile 0..15
    const int hi   = lane >> 4;             // which lane half
    const int m    = (wave << 4) + (lane & 15);

    v8f acc = {};
    #pragma unroll
    for (int k0 = 0; k0 < HH; k0 += 4) {
        const int k = k0 + (hi << 1);       // this lane's K base: VGPR j -> K=k+j
        v2f a, bv;
        a[0]  = W1[m * HH + k];
        a[1]  = W1[m * HH + k + 1];
        bv[0] = W2[k];                      // broadcast over all N columns
        bv[1] = W2[k + 1];
        acc = __builtin_amdgcn_wmma_f32_16x16x4_f32(
            /*neg_a=*/false, a, /*neg_b=*/false, bv,
            /*c_mod=*/(short)0, acc, /*reuse_a=*/false, /*reuse_b=*/false);
    }

    if ((lane & 15) == 0) {                 // lanes 0 and 16: column N=0
        const int mbase = (wave << 4) + (hi << 3);
        #pragma unroll
        for (int r = 0; r < 8; ++r) wc[mbase + r] = acc[r];
    }
    if (threadIdx.x == 0) {
        float s = 0.f;
        for (int h = 0; h < HH; ++h) s += b1[h] * W2[h];
        wc[DD] = (float)FF * (s + b2[0]);
    }
}

// ---------------------------------------------------------------------------
// Kernel 2 (bandwidth-bound main path):
//   out[b] = c + sum_f dot(tables[f, src[b,f], :], w)
// One wave32 per b. Lane l owns row elements [8l, 8l+8): two b128 loads per
// lane cover the 1KB row per wave, fully coalesced. w lives in 8 VGPRs/lane.
// Indices are wave-uniform -> scalar loads. Next row is prefetched
// (global_prefetch_b8) to hide random-gather latency.
// ---------------------------------------------------------------------------
__global__ __launch_bounds__(256) void gather_dot_kernel(
    const int* __restrict__ src, const float* __restrict__ tables,
    const float* __restrict__ wc, float* __restrict__ out)
{
    const int lane = threadIdx.x & 31;
    const int wave = threadIdx.x >> 5;
    const int b    = (blockIdx.x << 3) + wave;

    const float4 w0 = *(const float4*)(wc + lane * 8);
    const float4 w1 = *(const float4*)(wc + lane * 8 + 4);
    const float  cb = wc[DD];

    const int* __restrict__ sb = src + b * FF;
    float acc = 0.f;
    int idx = sb[0];

    for (int f = 0; f < FF; ++f) {
        const float* __restrict__ row =
            tables + ((size_t)f * VV + (size_t)idx) * DD + lane * 8;
        if (f + 1 < FF) {
            const int nidx = sb[f + 1];
            const float* nrow =
                tables + ((size_t)(f + 1) * VV + (size_t)nidx) * DD + lane * 8;
            __builtin_prefetch(nrow, 0, 3);
            idx = nidx;
        }
        const float4 r0 = *(const float4*)(row);
        const float4 r1 = *(const float4*)(row + 4);
        acc = fmaf(r0.x, w0.x, acc);
        acc = fmaf(r0.y, w0.y, acc);
        acc = fmaf(r0.z, w0.z, acc);
        acc = fmaf(r0.w, w0.w, acc);
        acc = fmaf(r1.x, w1.x, acc);
        acc = fmaf(r1.y, w1.y, acc);
        acc = fmaf(r1.z, w1.z, acc);
        acc = fmaf(r1.w, w1.w, acc);
    }

    #pragma unroll
    for (int off = 16; off > 0; off >>= 1)
        acc += __shfl_xor(acc, off, 32);
    if (lane == 0) out[b] = acc + cb;
}

// ---------------------------------------------------------------------------
extern "C" void kernel_launch(void* const* d_in, const int* in_sizes, int n_in,
                              void* d_out, int out_size, void* d_ws, size_t ws_size,
                              hipStream_t stream) {
    const int*   src    = (const int*)  d_in[0];
    const float* tables = (const float*)d_in[1];
    const float* W1     = (const float*)d_in[2];
    const float* b1     = (const float*)d_in[3];
    const float* W2     = (const float*)d_in[4];
    const float* b2     = (const float*)d_in[5];
    float* out = (float*)d_out;
    float* wc  = (float*)d_ws;          // 257 floats of scratch

    fold_mlp_kernel<<<1, 512, 0, stream>>>(W1, b1, W2, b2, wc);
    gather_dot_kernel<<<BB / 8, 256, 0, stream>>>(src, tables, wc, out);
}